// ModuleUsingCifHr_82454782148717
// MI455X (gfx1250) — compile-verified
//
#include <hip/hip_runtime.h>
#include <hip/hip_bf16.h>

// ---- Problem constants (from the reference) ----
#define STRIDE_   8
#define THRESH_   0.1f
#define INV_NEIGH (1.0f / 16.0f)   // FACTOR / NEIGHBORS
#define K_WIN     15
#define F_OUT     17               // x[1:] -> 17 fields
#define H_IN      80
#define W_IN      80
#define HH        633              // (80-1)*8+1
#define WH        633

typedef __attribute__((ext_vector_type(2))) float v2f;
typedef __attribute__((ext_vector_type(8))) float v8f;

__global__ void cifhr_zero_kernel(float* __restrict__ out, int n) {
  int i = blockIdx.x * blockDim.x + threadIdx.x;
  int stride = gridDim.x * blockDim.x;
  for (; i < n; i += stride) out[i] = 0.0f;
}

__global__ void cifhr_clamp_kernel(float* __restrict__ out, int n) {
  int i = blockIdx.x * blockDim.x + threadIdx.x;
  int stride = gridDim.x * blockDim.x;
  for (; i < n; i += stride) out[i] = fminf(out[i], 1.0f);
}

// One wave32 per source point. The 15x15 Gaussian patch is the rank-1 outer
// product wy (rows, vv0 folded in) x wx (cols), computed on the matrix pipe
// with a single V_WMMA_F32_16X16X4_F32 (K columns 1..3 zeroed). Masked /
// padded entries are exactly zero, so their atomics are skipped.
__global__ void cifhr_splat_kernel(const float* __restrict__ x,
                                   float* __restrict__ out) {
  const int lane = threadIdx.x & 31;
  const int wave = threadIdx.x >> 5;
  const int p = blockIdx.x * (blockDim.x >> 5) + wave;
  const int npoints = F_OUT * H_IN * W_IN;
  if (p >= npoints) return;  // wave-uniform (p identical across the wave)

  const int f   = p / (H_IN * W_IN);
  const int rem = p - f * (H_IN * W_IN);
  const int py  = rem / W_IN;
  const int px  = rem - py * W_IN;

  // x layout: (18, 5, 80, 80); field f is batch index f+1.
  const size_t hw   = (size_t)H_IN * W_IN;
  const size_t base = ((size_t)(f + 1) * 5) * hw + (size_t)py * W_IN + px;
  const float v  = x[base + 1 * hw];   // conf
  const float rx = x[base + 2 * hw];
  const float ry = x[base + 3 * hw];
  const float sc = x[base + 4 * hw];   // scale

  // keep = (v > THRESHOLD) & (scale >= MIN_SCALE/STRIDE == 0)
  int keep = (v > THRESH_) && (sc >= 0.0f);
  keep = __builtin_amdgcn_readfirstlane(keep);  // scalar branch: EXEC stays all-1s
  if (!keep) return;

  const float cx     = rx * (float)STRIDE_;
  const float cy     = ry * (float)STRIDE_;
  const float sigma  = fmaxf(1.0f, 0.5f * sc * (float)STRIDE_);
  const float inv2s2 = 0.5f / (sigma * sigma);
  const float vv0    = v * INV_NEIGH;

  const int ix_lo = (int)floorf(cx - sigma);
  const int iy_lo = (int)floorf(cy - sigma);
  const int ix_hi = (int)floorf(cx + sigma + 1.0f);
  const int iy_hi = (int)floorf(cy + sigma + 1.0f);

  // Lanes 0..15 build the row weights wy[m] (A, K=0) and column weights wx[n]
  // (B, K=0). 15->16 pad and all boundary masks are zeros in the weights.
  const int   idxk = lane & 15;
  const int   jjw  = iy_lo + idxk;
  const int   iiw  = ix_lo + idxk;
  const bool  inA  = (lane < 16) && (idxk < K_WIN);
  const bool  my   = inA && (jjw >= 0) && (jjw < HH) && (jjw < iy_hi);
  const bool  mx   = inA && (iiw >= 0) && (iiw < WH) && (iiw < ix_hi);
  const float dy   = (float)jjw - cy;
  const float dx   = (float)iiw - cx;
  const float ey   = __expf(-dy * dy * inv2s2) * vv0;
  const float ex   = __expf(-dx * dx * inv2s2);
  const float wy   = my ? ey : 0.0f;
  const float wx   = mx ? ex : 0.0f;

  v2f a = {wy, 0.0f};   // A 16x4 f32: VGPR0 lanes 0-15 = (M=lane, K=0)
  v2f b = {wx, 0.0f};   // B 4x16 f32: VGPR0 lanes 0-15 = (K=0, N=lane)
  v8f c = {};
  // D[m][n] = wy[m] * wx[n]  (rank-1 outer product on the matrix pipe)
  v8f d = __builtin_amdgcn_wmma_f32_16x16x4_f32(
      /*neg_a=*/false, a, /*neg_b=*/false, b,
      /*c_mod=*/(short)0, c, /*reuse_a=*/false, /*reuse_b=*/false);

  // D layout: VGPR r, lanes 0-15 -> M=r, N=lane; lanes 16-31 -> M=r+8, N=lane-16.
  const int half = lane >> 4;
  const int col  = lane & 15;
  const int ii   = ix_lo + col;
  float* outf = out + (size_t)f * HH * WH;
#pragma unroll
  for (int r = 0; r < 8; ++r) {
    const int m  = r + half * 8;
    const int jj = iy_lo + m;
    const float val = d[r];
    if (val != 0.0f) {  // zero iff masked out-of-bounds -> indices valid here
      atomicAdd(outf + (size_t)jj * WH + ii, val);
    }
  }
}

extern "C" void kernel_launch(void* const* d_in, const int* in_sizes, int n_in,
                              void* d_out, int out_size, void* d_ws, size_t ws_size,
                              hipStream_t stream) {
  (void)in_sizes; (void)n_in; (void)d_ws; (void)ws_size;
  const float* x = (const float*)d_in[0];
  float* out = (float*)d_out;

  // 1) zero the accumulation grid (d_out is poisoned before timing)
  cifhr_zero_kernel<<<2048, 256, 0, stream>>>(out, out_size);

  // 2) splat: one wave32 per source point, 8 waves per 256-thread block
  const int npoints = F_OUT * H_IN * W_IN;      // 108800
  const int waves_per_block = 8;
  const int blocks = (npoints + waves_per_block - 1) / waves_per_block;
  cifhr_splat_kernel<<<blocks, waves_per_block * 32, 0, stream>>>(x, out);

  // 3) clamp to 1.0 in place
  cifhr_clamp_kernel<<<2048, 256, 0, stream>>>(out, out_size);
}